// Attention_45810121179566
// MI455X (gfx1250) — compile-verified
//
#include <hip/hip_runtime.h>

typedef __attribute__((ext_vector_type(2))) float v2f;
typedef __attribute__((ext_vector_type(8))) float v8f;
typedef __attribute__((ext_vector_type(4))) unsigned int v4u;
typedef __attribute__((ext_vector_type(8))) int v8i;
typedef __attribute__((ext_vector_type(4))) int v4i;

#define BATCH 128
#define LEN   256
#define CH    256
#define NH    8
#define HD    32
#define QKV_LD 768
#define SP    260   // LDS pitch for 256-wide P rows (bank-conflict free, covers prefetch overread)

__device__ __forceinline__ v8f wmma4(v2f a, v2f b, v8f c) {
  // D = A(16x4,f32) * B(4x16,f32) + C(16x16,f32)
  return __builtin_amdgcn_wmma_f32_16x16x4_f32(false, a, false, b, (short)0, c,
                                               false, false);
}

// ---------------------------------------------------------------------------
// M x 256 x N GEMM + bias. W strip (256 rows x 64 cols, row stride N) is
// DMA'd into 64KB dynamic LDS by the Tensor Data Mover (one tensor_load_to_lds
// issued by wave 0; TENSORcnt + barrier for visibility). Each wave computes
// one 16-row tile x 64-col strip; A prefetched one K-step ahead; B fragments
// grouped so WMMAs issue back-to-back.
// grid.x = Mtiles/8, grid.y = N/64, block = 256 (8 waves), dynLDS = 64KB.
// ---------------------------------------------------------------------------
__global__ __launch_bounds__(256) void gemm_bias_kernel(
    const float* __restrict__ A, const float* __restrict__ W,
    const float* __restrict__ bias, float* __restrict__ out, int N) {
  extern __shared__ float sW[];  // [256][64], bank = n -> conflict free
  const int lane   = threadIdx.x & 31;
  const int wave   = threadIdx.x >> 5;
  const int m0     = (blockIdx.x * 8 + wave) * 16;
  const int n0     = blockIdx.y * 64;
  const int laneLo = lane & 15;
  const int koff   = (lane >> 4) << 1;

  // ---- TDM: DMA the K x 64 weight strip (row stride N) into LDS ----
  if (threadIdx.x < 32) {
    const unsigned long long ga = (unsigned long long)(const void*)(W + n0);
    v4u g0;
    g0.x = 1u;                              // count=1 (valid descriptor)
    g0.y = 0u;                              // lds_addr = 0 (start of dyn LDS)
    g0.z = (unsigned int)ga;                // global_addr[31:0]
    g0.w = (unsigned int)(ga >> 32) | 0x80000000u;  // addr[56:32] | type=2
    v8i g1;
    g1[0] = 0x00020000;                     // data_size=2 (4B), no mask/pad
    g1[1] = 64 << 16;                       // tensor_dim0 = 64
    g1[2] = 256 << 16;                      // tensor_dim1 = 256
    g1[3] = 64 << 16;                       // tile_dim0 = 64
    g1[4] = 256;                            // tile_dim1 = 256 (tile_dim2=0)
    g1[5] = N;                              // tensor_dim0_stride = N elements
    g1[6] = 0;                              // stride hi / dim1_stride lo
    g1[7] = 0;
    v4i gz4 = {0, 0, 0, 0};                 // groups 2/3 unused (2D tensor)
    v8i gz8 = {0, 0, 0, 0, 0, 0, 0, 0};     // extra group (zero-filled)
    __builtin_amdgcn_tensor_load_to_lds(g0, g1, gz4, gz4, gz8, 0);
    __builtin_amdgcn_s_wait_tensorcnt(0);
  }
  __syncthreads();

  const float* arow = A + (size_t)(m0 + laneLo) * CH + koff;  // lda = 256
  v8f acc[4] = {};

  v2f a;
  a.x = arow[0];
  a.y = arow[1];
  for (int k0 = 0; k0 < CH; k0 += 4) {
    // prefetch next A fragment (guarded: stay in-bounds)
    v2f an = a;
    if (k0 + 4 < CH) {
      an.x = arow[k0 + 4];
      an.y = arow[k0 + 5];
    }
    const float* bs = sW + (k0 + koff) * 64 + laneLo;
    v2f bf[4];
#pragma unroll
    for (int t = 0; t < 4; ++t) {
      bf[t].x = bs[t * 16];
      bf[t].y = bs[64 + t * 16];
    }
#pragma unroll
    for (int t = 0; t < 4; ++t) acc[t] = wmma4(a, bf[t], acc[t]);
    a = an;
  }

  const int rbase = m0 + ((lane >> 4) << 3);
#pragma unroll
  for (int t = 0; t < 4; ++t) {
    const float bv = bias[n0 + t * 16 + laneLo];
#pragma unroll
    for (int r = 0; r < 8; ++r)
      out[(size_t)(rbase + r) * N + n0 + t * 16 + laneLo] = acc[t][r] + bv;
  }
}

// ---------------------------------------------------------------------------
// LePE: depthwise 3x3 SAME conv on V (viewed as (B, C, 16, 16)); result
// written into ctx (B, L, C) — also initializes ctx for PV accumulation.
// grid.x = B*L, block = 256 (one thread per channel).
// ---------------------------------------------------------------------------
__global__ __launch_bounds__(256) void lepe_kernel(
    const float* __restrict__ qkv, const float* __restrict__ conv_w,
    const float* __restrict__ conv_b, float* __restrict__ ctx) {
  const int b = blockIdx.x >> 8;
  const int l = blockIdx.x & 255;
  const int c = threadIdx.x;
  const int y = l >> 4, x = l & 15;
  float acc = conv_b[c];
#pragma unroll
  for (int ky = 0; ky < 3; ++ky) {
    const int yy = y + ky - 1;
    if (yy < 0 || yy > 15) continue;
#pragma unroll
    for (int kx = 0; kx < 3; ++kx) {
      const int xx = x + kx - 1;
      if (xx < 0 || xx > 15) continue;
      acc += conv_w[c * 9 + ky * 3 + kx] *
             qkv[((size_t)b * LEN + (yy * 16 + xx)) * QKV_LD + 512 + c];
    }
  }
  ctx[((size_t)b * LEN + l) * CH + c] = acc;
}

// ---------------------------------------------------------------------------
// Fused attention: per (128-row q-block, head, batch) workgroup.
// S = scale*Q*K^T + mask -> LDS; 2-thread-per-row softmax; attn written once;
// O = P*V accumulated into ctx (which already holds LePE).
// grid = (2, 8, 128), block = 256 (8 waves), dynLDS = 128*260*4 + 1KB.
// ---------------------------------------------------------------------------
__global__ __launch_bounds__(256) void attn_kernel(
    const float* __restrict__ qkv, const float* __restrict__ mask,
    float* __restrict__ attn_out, float* __restrict__ ctx) {
  extern __shared__ float sP[];          // [128][SP] + red[256]
  float* red = sP + 128 * SP;            // softmax reduction scratch
  const int qblk = blockIdx.x;   // 0..1
  const int h    = blockIdx.y;   // 0..7
  const int b    = blockIdx.z;   // 0..127
  const int lane   = threadIdx.x & 31;
  const int wave   = threadIdx.x >> 5;
  const int laneLo = lane & 15;
  const int koff   = (lane >> 4) << 1;

  const float* qptr = qkv + (size_t)b * LEN * QKV_LD + h * HD;
  const float* kptr = qptr + 256;
  const float* vptr = qptr + 512;
  const int qr = qblk * 128 + wave * 16;  // global q-row base of this wave

  // ---- Phase 1: S strip (16 rows x 256 cols) via WMMA, K = hd = 32 ----
  v8f acc[16] = {};
  const float* aq = qptr + (size_t)(qr + laneLo) * QKV_LD + koff;
  v2f a;
  a.x = aq[0];
  a.y = aq[1];
  for (int k0 = 0; k0 < HD; k0 += 4) {
    v2f an = a;
    if (k0 + 4 < HD) {          // prefetch next A fragment
      an.x = aq[k0 + 4];
      an.y = aq[k0 + 5];
    }
    v2f bf[16];
#pragma unroll
    for (int nt = 0; nt < 16; ++nt) {
      const float* kb = kptr + (size_t)(nt * 16 + laneLo) * QKV_LD + k0 + koff;
      bf[nt].x = kb[0];
      bf[nt].y = kb[1];
    }
#pragma unroll
    for (int nt = 0; nt < 16; ++nt) acc[nt] = wmma4(a, bf[nt], acc[nt]);
    a = an;
  }

  const float scale = 0.1767766952966369f;  // 1/sqrt(32)
  const float* mrow = mask + (size_t)(b & 63) * LEN * LEN;
  const int lrbase = wave * 16 + ((lane >> 4) << 3);
#pragma unroll
  for (int nt = 0; nt < 16; ++nt) {
#pragma unroll
    for (int r = 0; r < 8; ++r) {
      const int lr  = lrbase + r;
      const int col = nt * 16 + laneLo;
      sP[lr * SP + col] =
          acc[nt][r] * scale + mrow[(size_t)(qblk * 128 + lr) * LEN + col];
    }
  }
  __syncthreads();

  // ---- Phase 2: softmax, 2 threads per row (128 cols each) ----
  {
    const int row  = threadIdx.x & 127;
    const int half = threadIdx.x >> 7;          // 0 or 1
    float* rp = sP + row * SP + half * 128;
    float m = rp[0];
    for (int j = 1; j < 128; ++j) m = fmaxf(m, rp[j]);
    red[half * 128 + row] = m;
    __syncthreads();
    m = fmaxf(red[row], red[128 + row]);
    float s = 0.f;
    for (int j = 0; j < 128; ++j) {
      const float e = __expf(rp[j] - m);
      rp[j] = e;
      s += e;
    }
    __syncthreads();                            // red reuse hazard
    red[half * 128 + row] = s;
    __syncthreads();
    const float inv = 1.0f / (red[row] + red[128 + row]);
    for (int j = 0; j < 128; ++j) rp[j] *= inv;
  }
  __syncthreads();

  // ---- Phase 3: write attn (coalesced, exactly once) ----
  float* aout = attn_out + ((size_t)(b * NH + h) * LEN + qblk * 128) * LEN;
  for (int i = threadIdx.x; i < 128 * LEN; i += 256) {
    const int r = i >> 8, cc = i & 255;
    aout[(size_t)r * LEN + cc] = sP[r * SP + cc];
  }

  // ---- Phase 4: O = P * V (K = 256), accumulate into ctx (+= LePE) ----
  v8f oacc[2] = {};
  const float* ap = sP + (wave * 16 + laneLo) * SP + koff;
  v2f pa;
  pa.x = ap[0];
  pa.y = ap[1];
  for (int k0 = 0; k0 < LEN; k0 += 4) {
    v2f pn;
    pn.x = ap[k0 + 4];   // SP=260 pad makes the last-step overread safe
    pn.y = ap[k0 + 5];
    v2f bf[2];
#pragma unroll
    for (int nt = 0; nt < 2; ++nt) {
      const float* vb = vptr + (size_t)(k0 + koff) * QKV_LD + nt * 16 + laneLo;
      bf[nt].x = vb[0];
      bf[nt].y = vb[QKV_LD];
    }
#pragma unroll
    for (int nt = 0; nt < 2; ++nt) oacc[nt] = wmma4(pa, bf[nt], oacc[nt]);
    pa = pn;
  }
#pragma unroll
  for (int nt = 0; nt < 2; ++nt) {
#pragma unroll
    for (int r = 0; r < 8; ++r) {
      float* p = ctx + ((size_t)b * LEN + qblk * 128 + lrbase + r) * CH +
                 h * HD + nt * 16 + laneLo;
      *p += oacc[nt][r];
    }
  }
}

// ---------------------------------------------------------------------------
extern "C" void kernel_launch(void* const* d_in, const int* in_sizes, int n_in,
                              void* d_out, int out_size, void* d_ws,
                              size_t ws_size, hipStream_t stream) {
  const float* x      = (const float*)d_in[0];
  const float* mask   = (const float*)d_in[1];
  const float* w_qkv  = (const float*)d_in[2];
  const float* b_qkv  = (const float*)d_in[3];
  const float* conv_w = (const float*)d_in[4];
  const float* conv_b = (const float*)d_in[5];
  const float* w_proj = (const float*)d_in[6];
  const float* b_proj = (const float*)d_in[7];

  float* out  = (float*)d_out;                       // (128,256,256)
  float* attn = out + (size_t)BATCH * LEN * CH;      // (128,8,256,256)
  float* qkv  = (float*)d_ws;                        // (32768,768)
  float* ctx  = qkv + (size_t)BATCH * LEN * 3 * CH;  // (32768,256)

  const size_t gemmLds = 256 * 64 * sizeof(float);               // 64 KB
  const size_t attnLds = (128 * SP + 256) * sizeof(float);       // ~131 KB

  // 1) qkv = x @ w_qkv + b_qkv       (M=32768, K=256, N=768)
  gemm_bias_kernel<<<dim3(256, 12), 256, gemmLds, stream>>>(x, w_qkv, b_qkv,
                                                            qkv, 768);
  // 2) ctx = LePE(v)                 (depthwise 3x3 + bias)
  lepe_kernel<<<dim3(BATCH * LEN), 256, 0, stream>>>(qkv, conv_w, conv_b, ctx);
  // 3) attn = softmax(scale*QK^T + mask); ctx += attn @ V
  attn_kernel<<<dim3(2, NH, BATCH), 256, attnLds, stream>>>(qkv, mask, attn,
                                                            ctx);
  // 4) out = ctx @ w_proj + b_proj   (M=32768, K=256, N=256)
  gemm_bias_kernel<<<dim3(256, 4), 256, gemmLds, stream>>>(ctx, w_proj, b_proj,
                                                           out, 256);
}